// Network_18382460027228
// MI455X (gfx1250) — compile-verified
//
#include <hip/hip_runtime.h>

typedef __attribute__((ext_vector_type(2))) float v2f;
typedef __attribute__((ext_vector_type(8))) float v8f;

#define T_STEPS 10
#define T_PAD   12        // K padded to 3 chunks of 4 for WMMA f32 16x16x4
#define F_DIM   16384
#define FB      1024      // f-columns per block
#define YROW    1032      // LDS row stride (floats); 2*YROW % 64 == 16 -> conflict-free half-wave reads

struct alignas(16) F4 { float v[4]; };

__global__ __launch_bounds__(256) void lif_net_kernel(
    const float* __restrict__ x,
    const float* __restrict__ tau1, const float* __restrict__ tau2, const float* __restrict__ tau3,
    const float* __restrict__ vth1p, const float* __restrict__ vth2p, const float* __restrict__ vth3p,
    const float* __restrict__ conv_w, const float* __restrict__ conv_b,
    const float* __restrict__ lin_w, const float* __restrict__ lin_b,
    float* __restrict__ out)
{
    __shared__ float lds_y[T_PAD * YROW];   // y[t][f_local], t rows 10,11 zeroed (K padding)
    __shared__ float lds_a[16 * T_PAD];     // lin_w padded to 16 x 12 (rows >=2 and t >= 10 are 0)

    const int tid  = threadIdx.x;
    const int b    = blockIdx.y;
    const int fblk = blockIdx.x * FB;

    // ---- stage padded A matrix (lin_w) into LDS ----
    for (int idx = tid; idx < 16 * T_PAD; idx += 256) {
        const int r = idx / T_PAD, c = idx % T_PAD;
        lds_a[idx] = (r < 2 && c < T_STEPS) ? lin_w[r * T_STEPS + c] : 0.0f;
    }

    // ---- phase 1: fused triple-LIF scan + channel mix, streamed once ----
    const float k1 = 0.001f * tau1[0];
    const float k2 = 0.001f * tau2[0];
    const float k3 = 0.001f * tau3[0];
    const float th1 = vth1p[0], th2 = vth2p[0], th3 = vth3p[0];
    const float cw0 = conv_w[0], cw1 = conv_w[1], cw2 = conv_w[2], cb = conv_b[0];

    const int f0 = fblk + tid * 4;
    float v1[4] = {0.f,0.f,0.f,0.f};
    float v2a[4] = {0.f,0.f,0.f,0.f};
    float v3[4] = {0.f,0.f,0.f,0.f};

    #pragma unroll
    for (int t = 0; t < T_STEPS; ++t) {
        const float* px = x + ((size_t)b * T_STEPS + t) * F_DIM + f0;
        if (t + 1 < T_STEPS)
            __builtin_prefetch(x + ((size_t)b * T_STEPS + t + 1) * F_DIM + f0, 0, 0);
        F4 xv = *(const F4*)px;     // global_load_b128, coalesced across the wave
        F4 yv;
        #pragma unroll
        for (int i = 0; i < 4; ++i) {
            float u1 = v1[i]  + k1 * (xv.v[i] - v1[i]);
            float z1 = (u1 > th1) ? 1.0f : 0.0f;
            v1[i]    = (u1 > th1) ? 0.0f : u1;
            float u2 = v2a[i] + k2 * (xv.v[i] - v2a[i]);
            float z2 = (u2 > th2) ? 1.0f : 0.0f;
            v2a[i]   = (u2 > th2) ? 0.0f : u2;
            float u3 = v3[i]  + k3 * (xv.v[i] - v3[i]);
            float z3 = (u3 > th3) ? 1.0f : 0.0f;
            v3[i]    = (u3 > th3) ? 0.0f : u3;
            yv.v[i]  = cw0 * z1 + cw1 * z2 + cw2 * z3 + cb;
        }
        *(F4*)(&lds_y[t * YROW + tid * 4]) = yv;   // ds_store_b128
    }
    {   // zero K-padding rows t = 10, 11
        F4 zf = {};
        *(F4*)(&lds_y[10 * YROW + tid * 4]) = zf;
        *(F4*)(&lds_y[11 * YROW + tid * 4]) = zf;
    }
    __syncthreads();

    // ---- phase 2: time-axis linear (2x10 @ 10xF) via V_WMMA_F32_16X16X4_F32 ----
    const int lane  = tid & 31;
    const int wave  = tid >> 5;
    const int m     = lane & 15;
    const int khalf = (lane >> 4) * 2;   // lanes 0-15 hold K=0,1; lanes 16-31 hold K=2,3

    v2f afrag[3];
    #pragma unroll
    for (int kc = 0; kc < 3; ++kc) {
        const int t0 = 4 * kc + khalf;
        afrag[kc][0] = lds_a[m * T_PAD + t0];
        afrag[kc][1] = lds_a[m * T_PAD + t0 + 1];
    }
    const float lb0 = lin_b[0], lb1 = lin_b[1];

    #pragma unroll
    for (int j = 0; j < 8; ++j) {
        const int tb = wave * 128 + j * 16;      // 16-column f tile within block
        v8f acc = {0.f,0.f,0.f,0.f,0.f,0.f,0.f,0.f};
        #pragma unroll
        for (int kc = 0; kc < 3; ++kc) {
            const int t0 = 4 * kc + khalf;
            v2f bfrag;
            bfrag[0] = lds_y[t0 * YROW + tb + m];
            bfrag[1] = lds_y[(t0 + 1) * YROW + tb + m];
            acc = __builtin_amdgcn_wmma_f32_16x16x4_f32(
                false, afrag[kc], false, bfrag, (short)0, acc, false, false);
        }
        // D layout: VGPR r, lanes 0-15 => row M=r. We need rows 0 and 1 only.
        if (lane < 16) {
            const int f = fblk + tb + lane;
            out[(b * 2 + 0) * F_DIM + f] = acc[0] + lb0;
            out[(b * 2 + 1) * F_DIM + f] = acc[1] + lb1;
        }
    }
}

extern "C" void kernel_launch(void* const* d_in, const int* in_sizes, int n_in,
                              void* d_out, int out_size, void* d_ws, size_t ws_size,
                              hipStream_t stream) {
    const float* x      = (const float*)d_in[0];
    const float* tau1   = (const float*)d_in[1];
    const float* tau2   = (const float*)d_in[2];
    const float* tau3   = (const float*)d_in[3];
    const float* vth1   = (const float*)d_in[4];
    const float* vth2   = (const float*)d_in[5];
    const float* vth3   = (const float*)d_in[6];
    const float* conv_w = (const float*)d_in[7];
    const float* conv_b = (const float*)d_in[8];
    const float* lin_w  = (const float*)d_in[9];
    const float* lin_b  = (const float*)d_in[10];
    float* out = (float*)d_out;

    const int Bsz = in_sizes[0] / (T_STEPS * F_DIM);   // 256
    dim3 grid(F_DIM / FB, Bsz);
    dim3 block(256);
    lif_net_kernel<<<grid, block, 0, stream>>>(
        x, tau1, tau2, tau3, vth1, vth2, vth3, conv_w, conv_b, lin_w, lin_b, out);
}